// GCN_Net_25769803776776
// MI455X (gfx1250) — compile-verified
//
#include <hip/hip_runtime.h>
#include <math.h>

#define NNODES 100000
#define NEDGES 1600000
#define FDIM   128
#define NCLS   40

typedef float v2f __attribute__((ext_vector_type(2)));
typedef float v8f __attribute__((ext_vector_type(8)));

// ---------------------------------------------------------------- degree prep
__global__ __launch_bounds__(256) void k_fill1(float* __restrict__ p, int n) {
    int i = blockIdx.x * 256 + threadIdx.x;
    if (i < n) p[i] = 1.0f;                       // self-loop contribution
}

__global__ __launch_bounds__(256) void k_deg(const int* __restrict__ dst,
                                             float* __restrict__ deg, int e) {
    int i = blockIdx.x * 256 + threadIdx.x;
    if (i < e) unsafeAtomicAdd(&deg[dst[i]], 1.0f);
}

__global__ __launch_bounds__(256) void k_rsqrt(float* __restrict__ p, int n) {
    int i = blockIdx.x * 256 + threadIdx.x;
    if (i < n) p[i] = rsqrtf(p[i]);               // deg >= 1 always (self-loops)
}

// ---------------------------------------------------------------- WMMA GEMM
// H[N x NCOLS] = act(X)[N x 128] @ W[128 x NCOLS], f32 WMMA 16x16x4.
// W is staged once per block into LDS; one wave computes a 16-row strip.
// RELU fuses max(x,0) into the A-fragment load (next-layer activation).
template <int NCOLS, bool RELU>
__global__ __launch_bounds__(256) void k_gemm(const float* __restrict__ X,
                                              const float* __restrict__ W,
                                              float* __restrict__ H) {
    constexpr int NT  = (NCOLS + 15) / 16;
    constexpr int LDW = NT * 16;                  // LDS row stride (48 for NCOLS=40)
    __shared__ float Wl[FDIM * LDW];              // 64KB (128-col) / 24KB (40-col)

    // cooperative stage of W into LDS (all 8 waves participate, then barrier)
    for (int idx = threadIdx.x; idx < FDIM * NCOLS; idx += 256) {
        int r = idx / NCOLS, c = idx - r * NCOLS;
        Wl[r * LDW + c] = W[idx];
    }
    __syncthreads();

    const int lane    = threadIdx.x & 31;
    const int rowtile = blockIdx.x * 8 + (threadIdx.x >> 5);
    if (rowtile >= NNODES / 16) return;           // wave-uniform guard (post-barrier)
    const int half = lane >> 4;                   // 0: lanes 0-15, 1: lanes 16-31
    const int lm   = lane & 15;

    const float* xrow = X + (long)(rowtile * 16 + lm) * FDIM;

    v8f acc[NT];
#pragma unroll
    for (int t = 0; t < NT; ++t) acc[t] = (v8f)0.0f;

#pragma unroll 4
    for (int k = 0; k < FDIM; k += 4) {
        const int kb = k + 2 * half;              // this lane's K rows: kb, kb+1
        v2f a = *(const v2f*)(xrow + kb);         // A frag: M=lm, K={kb,kb+1}
        if (RELU) { a.x = fmaxf(a.x, 0.0f); a.y = fmaxf(a.y, 0.0f); }
#pragma unroll
        for (int t = 0; t < NT; ++t) {
            const int col = t * 16 + lm;
            v2f b;
            b.x = Wl[kb * LDW + col];             // B frag: K=kb..kb+1, N=col
            b.y = Wl[(kb + 1) * LDW + col];
            if (NCOLS % 16 != 0 && col >= NCOLS) { b.x = 0.0f; b.y = 0.0f; }
            acc[t] = __builtin_amdgcn_wmma_f32_16x16x4_f32(
                false, a, false, b, (short)0, acc[t], false, false);
        }
    }

    // C/D layout: VGPR r -> M = r + 8*half, N = lm (per half)
#pragma unroll
    for (int t = 0; t < NT; ++t) {
        const int col = t * 16 + lm;
        if (NCOLS % 16 == 0 || col < NCOLS) {
#pragma unroll
            for (int r = 0; r < 8; ++r) {
                const long m = (long)rowtile * 16 + r + 8 * half;
                H[m * NCOLS + col] = acc[t][r];
            }
        }
    }
}

// --------------------------------------------- self-loop + bias initialization
// out[i,f] = h[i,f] * dis[i]^2 + b[f]
__global__ __launch_bounds__(256) void k_self_bias(const float* __restrict__ h,
                                                   const float* __restrict__ dis,
                                                   const float* __restrict__ b,
                                                   float* __restrict__ out,
                                                   int feats, long total) {
    long i = (long)blockIdx.x * 256 + threadIdx.x;
    if (i >= total) return;
    int node = (int)(i / feats);
    int f    = (int)(i - (long)node * feats);
    float d  = dis[node];
    out[i] = h[i] * d * d + b[f];
}

// ---------------------------------------------------------------- edge scatter
// one wave per edge; 128 feats -> float4 per lane, 4 f32 atomics
__global__ __launch_bounds__(256) void k_scatter128(const float* __restrict__ h,
                                                    const int* __restrict__ src,
                                                    const int* __restrict__ dst,
                                                    const float* __restrict__ dis,
                                                    float* __restrict__ out) {
    int e    = (blockIdx.x * 256 + threadIdx.x) >> 5;
    int lane = threadIdx.x & 31;
    if (e >= NEDGES) return;
    int s = src[e], d = dst[e];
    float norm = dis[s] * dis[d];
    float4 v = ((const float4*)(h + (long)s * FDIM))[lane];
    float* o = out + (long)d * FDIM + lane * 4;
    unsafeAtomicAdd(o + 0, v.x * norm);
    unsafeAtomicAdd(o + 1, v.y * norm);
    unsafeAtomicAdd(o + 2, v.z * norm);
    unsafeAtomicAdd(o + 3, v.w * norm);
}

// one wave per edge; 40 feats -> lane c and (lanes 0-7) c+32
__global__ __launch_bounds__(256) void k_scatter40(const float* __restrict__ h,
                                                   const int* __restrict__ src,
                                                   const int* __restrict__ dst,
                                                   const float* __restrict__ dis,
                                                   float* __restrict__ out) {
    int e    = (blockIdx.x * 256 + threadIdx.x) >> 5;
    int lane = threadIdx.x & 31;
    if (e >= NEDGES) return;
    int s = src[e], d = dst[e];
    float norm = dis[s] * dis[d];
    const float* hs = h + (long)s * NCLS;
    float* o = out + (long)d * NCLS;
    unsafeAtomicAdd(o + lane, hs[lane] * norm);
    if (lane < NCLS - 32) unsafeAtomicAdd(o + 32 + lane, hs[32 + lane] * norm);
}

// ---------------------------------------------------------------- log_softmax
// one wave per row of 40 classes; shuffle reductions
__global__ __launch_bounds__(256) void k_logsoftmax(float* __restrict__ out) {
    int row  = (blockIdx.x * 256 + threadIdx.x) >> 5;
    int lane = threadIdx.x & 31;
    if (row >= NNODES) return;
    float* o = out + (long)row * NCLS;
    float v0 = o[lane];
    float v1 = (lane < NCLS - 32) ? o[32 + lane] : -3.402823466e38f;
    float m = fmaxf(v0, v1);
#pragma unroll
    for (int off = 16; off >= 1; off >>= 1) m = fmaxf(m, __shfl_xor(m, off, 32));
    float s = expf(v0 - m) + ((lane < NCLS - 32) ? expf(v1 - m) : 0.0f);
#pragma unroll
    for (int off = 16; off >= 1; off >>= 1) s += __shfl_xor(s, off, 32);
    float lse = m + logf(s);
    o[lane] = v0 - lse;
    if (lane < NCLS - 32) o[32 + lane] = v1 - lse;
}

// ------------------------------------------------------------------- driver
extern "C" void kernel_launch(void* const* d_in, const int* in_sizes, int n_in,
                              void* d_out, int out_size, void* d_ws, size_t ws_size,
                              hipStream_t stream) {
    const float* x   = (const float*)d_in[0];
    const int*   eix = (const int*)d_in[1];      // [2, E] int32 (jax default x64-off)
    const float* W1  = (const float*)d_in[2];
    const float* b1  = (const float*)d_in[3];
    const float* W2  = (const float*)d_in[4];
    const float* b2  = (const float*)d_in[5];
    const float* W3  = (const float*)d_in[6];
    const float* b3  = (const float*)d_in[7];
    const int* src = eix;
    const int* dst = eix + NEDGES;

    float* wsf = (float*)d_ws;
    float* deg = wsf;                                  // N floats (becomes dis)
    float* h   = wsf + 102400;                         // N x 128 (reused as N x 40)
    float* agg = h + (size_t)NNODES * FDIM;            // N x 128
    float* out = (float*)d_out;                        // N x 40

    const dim3 blk(256);
    const int gN    = (NNODES + 255) / 256;
    const int gE    = (NEDGES + 255) / 256;
    const int gEw   = NEDGES * 32 / 256;               // one wave per edge
    const int gGemm = (NNODES / 16 + 7) / 8;           // 8 row-tiles per block
    const long t128 = (long)NNODES * FDIM;
    const long t40  = (long)NNODES * NCLS;
    const int g128  = (int)((t128 + 255) / 256);
    const int g40   = (int)((t40 + 255) / 256);
    const int gRow  = NNODES * 32 / 256 + 1;           // one wave per row

    // symmetric-normalization degrees
    k_fill1<<<gN, blk, 0, stream>>>(deg, NNODES);
    k_deg<<<gE, blk, 0, stream>>>(dst, deg, NEDGES);
    k_rsqrt<<<gN, blk, 0, stream>>>(deg, NNODES);

    // layer 1 (input x, no activation on A)
    k_gemm<128, false><<<gGemm, blk, 0, stream>>>(x, W1, h);
    k_self_bias<<<g128, blk, 0, stream>>>(h, deg, b1, agg, FDIM, t128);
    k_scatter128<<<gEw, blk, 0, stream>>>(h, src, dst, deg, agg);

    // layer 2 (ReLU fused into A-fragment load)
    k_gemm<128, true><<<gGemm, blk, 0, stream>>>(agg, W2, h);
    k_self_bias<<<g128, blk, 0, stream>>>(h, deg, b2, agg, FDIM, t128);
    k_scatter128<<<gEw, blk, 0, stream>>>(h, src, dst, deg, agg);

    // layer 3 (ReLU fused) + log_softmax
    k_gemm<40, true><<<gGemm, blk, 0, stream>>>(agg, W3, h);
    k_self_bias<<<g40, blk, 0, stream>>>(h, deg, b3, out, NCLS, t40);
    k_scatter40<<<gEw, blk, 0, stream>>>(h, src, dst, deg, out);
    k_logsoftmax<<<gRow, blk, 0, stream>>>(out);
}